// SpmmLinear_89833535963585
// MI455X (gfx1250) — compile-verified
//
#include <hip/hip_runtime.h>
#include <stdint.h>

typedef float v2f __attribute__((ext_vector_type(2)));
typedef float v8f __attribute__((ext_vector_type(8)));
typedef int   v4i __attribute__((ext_vector_type(4)));

#define BH    32
#define BW    32
#define LDSW  36      // padded LDS row stride in floats (144B: 16B-aligned, conflict-free)
#define WAVES 4
#define MTILE 128     // tokens per workgroup (32 per wave)

typedef __attribute__((address_space(1))) v4i g_v4i;   // global int4
typedef __attribute__((address_space(3))) v4i l_v4i;   // LDS int4

// ---- CDNA5 async global->LDS copy (16B per lane), tracked by ASYNCcnt ----
__device__ __forceinline__ void async_copy_b128(const void* gsrc, void* ldst) {
#if __has_builtin(__builtin_amdgcn_global_load_async_to_lds_b128)
    __builtin_amdgcn_global_load_async_to_lds_b128(
        (g_v4i*)gsrc, (l_v4i*)ldst, 0, 0);
#else
    uint32_t loff = (uint32_t)(uintptr_t)ldst;   // low 32 bits = LDS byte offset
    asm volatile("global_load_async_to_lds_b128 %0, %1, off"
                 :: "v"(loff), "v"(gsrc) : "memory");
#endif
}

__device__ __forceinline__ void wait_async0() {
#if __has_builtin(__builtin_amdgcn_s_wait_asynccnt)
    __builtin_amdgcn_s_wait_asynccnt(0);
#else
    asm volatile("s_wait_asynccnt 0" ::: "memory");
#endif
}

// Stage one sparse block: W (32x32, cooperative across 128 threads) and this
// wave's X tile (32 tokens x 32 feats) into LDS, fully async.
__device__ __forceinline__ void issue_tile_loads(const float* __restrict__ wgBlk,
                                                 const float* __restrict__ xg,
                                                 float* ldsW, float* ldsX,
                                                 int tid, int lane, int in_f) {
    // W block: 256 float4s over 128 threads -> 2 async b128 per thread
    const int i0 = tid, i1 = tid + 128;
    async_copy_b128(wgBlk + i0 * 4, &ldsW[(i0 >> 3) * LDSW + (i0 & 7) * 4]);
    async_copy_b128(wgBlk + i1 * 4, &ldsW[(i1 >> 3) * LDSW + (i1 & 7) * 4]);
    // X tile: 256 float4s over 32 lanes -> 8 async b128 per lane (coalesced rows)
    #pragma unroll
    for (int i = 0; i < 8; ++i) {
        const int idx = i * 32 + lane;
        const int r = idx >> 3, c4 = idx & 7;
        async_copy_b128(xg + (size_t)r * in_f + c4 * 4, &ldsX[r * LDSW + c4 * 4]);
    }
}

__global__ __launch_bounds__(128)
void spmm_block_wmma(const float* __restrict__ x,
                     const float* __restrict__ w,
                     const float* __restrict__ bias,
                     const int*   __restrict__ brow,
                     const int*   __restrict__ bcol,
                     float*       __restrict__ y,
                     int n_blocks, int in_f, int out_f)
{
    __shared__ float lds_w[2][BH * LDSW];            // 2 x 4.5 KB (double buffer)
    __shared__ float lds_x[2][WAVES][BH * LDSW];     // 2 x 18 KB

    const int tid  = threadIdx.x;
    const int wave = tid >> 5;
    const int lane = tid & 31;
    const int half = lane >> 4;
    const int l16  = lane & 15;

    const int row_blk  = blockIdx.x;                      // output 32-row block
    const int n_base   = blockIdx.y * MTILE + wave * 32;  // this wave's token base
    const int out_base = row_blk * BH;

    v8f acc[2][2] = {};   // 2x2 tiles of 16x16 f32 accumulators

    // find first matching sparse block (uniform scan -> scalar branches)
    int blk = 0;
    while (blk < n_blocks && brow[blk] != row_blk) ++blk;

    int cur = 0;
    if (blk < n_blocks) {
        issue_tile_loads(w + (size_t)blk * BH * BW,
                         x + (size_t)n_base * in_f + (size_t)bcol[blk] * BW,
                         lds_w[0], lds_x[0][wave], tid, lane, in_f);
    }

    while (blk < n_blocks) {
        // uniform scan for the next matching block
        int nxt = blk + 1;
        while (nxt < n_blocks && brow[nxt] != row_blk) ++nxt;

        // buffer `cur` ready once every wave drained its own ASYNCcnt + barrier
        wait_async0();
        __syncthreads();

        // overlap: DMA next block's tiles into the other buffer while we WMMA
        if (nxt < n_blocks) {
            __builtin_prefetch(w + (size_t)nxt * BH * BW + lane * 32, 0, 1);
            issue_tile_loads(w + (size_t)nxt * BH * BW,
                             x + (size_t)n_base * in_f + (size_t)bcol[nxt] * BW,
                             lds_w[cur ^ 1], lds_x[cur ^ 1][wave], tid, lane, in_f);
        }

        // --- 32x32x32 block GEMM: 8 k-steps x (2x2) v_wmma_f32_16x16x4_f32 ---
        const float* xt = lds_x[cur][wave];
        const float* wt = lds_w[cur];
        #pragma unroll
        for (int ks = 0; ks < 8; ++ks) {
            const int k0 = ks * 4;
            v2f a[2], b[2];
            #pragma unroll
            for (int t = 0; t < 2; ++t) {
                // A frag: A[m = l16 (+16t)][k = k0 + 2*half + v]   (ds_load_b64)
                a[t] = *(const v2f*)&xt[(t * 16 + l16) * LDSW + k0 + 2 * half];
                // B frag: B[k][n] = W[n][k]; n = l16 (+16t)        (ds_load_b64)
                b[t] = *(const v2f*)&wt[(t * 16 + l16) * LDSW + k0 + 2 * half];
            }
            #pragma unroll
            for (int tm = 0; tm < 2; ++tm)
                #pragma unroll
                for (int tn = 0; tn < 2; ++tn)
                    acc[tm][tn] = __builtin_amdgcn_wmma_f32_16x16x4_f32(
                        false, a[tm], false, b[tn],
                        (short)0, acc[tm][tn], false, false);
        }

        blk = nxt;
        cur ^= 1;
        // no trailing barrier: next iteration's wait_async0+barrier orders both
        // the async fills of the new `cur` and every wave's compute on it.
    }

    // --- epilogue: fuse bias, single plain store (no atomics needed) ---
    float bv[2];
    bv[0] = bias[out_base + l16];
    bv[1] = bias[out_base + 16 + l16];
    #pragma unroll
    for (int tm = 0; tm < 2; ++tm) {
        #pragma unroll
        for (int tn = 0; tn < 2; ++tn) {
            #pragma unroll
            for (int v = 0; v < 8; ++v) {
                const int m = tm * 16 + v + 8 * half;   // D row per ISA C/D layout
                y[(size_t)(n_base + m) * out_f + out_base + tn * 16 + l16]
                    = acc[tm][tn][v] + bv[tn];
            }
        }
    }
}

extern "C" void kernel_launch(void* const* d_in, const int* in_sizes, int n_in,
                              void* d_out, int out_size, void* d_ws, size_t ws_size,
                              hipStream_t stream) {
    const float* x    = (const float*)d_in[0];
    const float* w    = (const float*)d_in[1];
    const float* bias = (const float*)d_in[2];
    const int*   brow = (const int*)d_in[3];
    const int*   bcol = (const int*)d_in[4];
    float*       y    = (float*)d_out;

    const int out_f    = in_sizes[2];              // 1024
    const int n_blocks = in_sizes[3];              // 256
    const int n_tok    = out_size / out_f;         // 16384
    const int in_f     = in_sizes[0] / n_tok;      // 1024

    dim3 grid(out_f / BH, n_tok / MTILE);          // 32 x 128 workgroups
    dim3 block(128);                               // 4 waves
    spmm_block_wmma<<<grid, block, 0, stream>>>(x, w, bias, brow, bcol, y,
                                                n_blocks, in_f, out_f);
}